// AffineEdgeAttention_56049323213775
// MI455X (gfx1250) — compile-verified
//
#include <hip/hip_runtime.h>

typedef __attribute__((ext_vector_type(2))) float v2f;
typedef __attribute__((ext_vector_type(8))) float v8f;

#define SEQ 1024
#define DIM 768

// Phase 1: one wave per row. s_h[b,i] = dot(head[b,i,:], w_h),
// s_d[b,j] = dot(dep[b,j,:], w_d) + bias (bias folded in here).
__global__ __launch_bounds__(256) void AffineEdge_rowdot(
    const float* __restrict__ head, const float* __restrict__ dep,
    const float* __restrict__ W, const float* __restrict__ bptr,
    float* __restrict__ sH, float* __restrict__ sD)
{
    const int gtid   = blockIdx.x * 256 + threadIdx.x;
    const int wave   = gtid >> 5;            // 0 .. 32767
    const int lane   = threadIdx.x & 31;
    const int tensor = wave >> 14;           // 0 = head, 1 = dep
    const int row    = wave & 16383;         // b*1024 + s

    const float* __restrict__ src = tensor ? dep : head;
    const float4* __restrict__ rp = (const float4*)(src + (size_t)row * DIM);
    const float4* __restrict__ wp = (const float4*)(W + tensor * DIM);

    float acc = 0.f;
#pragma unroll
    for (int t = 0; t < 6; ++t) {            // 32 lanes * 6 * float4 = 768
        float4 a = rp[lane + 32 * t];
        float4 w = wp[lane + 32 * t];
        acc += a.x * w.x + a.y * w.y + a.z * w.z + a.w * w.w;
    }
#pragma unroll
    for (int off = 16; off > 0; off >>= 1)   // wave32 butterfly reduce
        acc += __shfl_xor(acc, off, 32);

    if (lane == 0) {
        if (tensor) sD[row] = acc + bptr[0];
        else        sH[row] = acc;
    }
}

// Phase 2: each wave owns a 16-row x 256-col strip of one batch image.
// Each 16x16 tile is produced by a single V_WMMA_F32_16X16X4_F32:
//   A(16x4): col K=0 = s_h[i0..i0+15], col K=1 = 1, K=2,3 = 0
//   B(4x16): row K=0 = 1,              row K=1 = s_d[j0..j0+15]+bias
//   D[i][j] = s_h[i] + s_d[j] + bias
__global__ __launch_bounds__(256) void AffineEdge_tile(
    const float* __restrict__ sH, const float* __restrict__ sD,
    float* __restrict__ out)
{
    const int gtid = blockIdx.x * 256 + threadIdx.x;
    const int wave = gtid >> 5;              // 0 .. 4095
    const int lane = threadIdx.x & 31;
    const int half = lane >> 4;              // 0: lanes 0-15, 1: lanes 16-31
    const int lid  = lane & 15;

    const int b      = wave >> 8;            // 16 batches
    const int itile  = (wave >> 2) & 63;     // 64 row-tiles
    const int jchunk = wave & 3;             // 4 column chunks of 256
    const int i0     = itile * 16;
    const int jbase  = jchunk * 256;

    // A operand: lanes 0-15 hold (K=0,K=1) = (s_h, 1); lanes 16-31 hold (K=2,K=3) = 0
    const float sh = sH[b * SEQ + i0 + lid];
    v2f A;
    A.x = half ? 0.f : sh;
    A.y = half ? 0.f : 1.f;

    float* __restrict__ ob = out + (size_t)b * SEQ * SEQ;
    // per-lane base: row (i0 + 8*half), col (jbase + lid); VGPR r adds r*SEQ floats
    float* p = ob + (size_t)(i0 + 8 * half) * SEQ + jbase + lid;
    const float* sdp = sD + b * SEQ + jbase + lid;

#pragma unroll 4
    for (int jt = 0; jt < 16; ++jt) {        // 16 tiles of 16 cols = 256 cols
        const float sd = sdp[jt * 16];
        v2f Bm;
        Bm.x = half ? 0.f : 1.f;             // K=0 row = ones
        Bm.y = half ? 0.f : sd;              // K=1 row = s_d + bias
        v8f C = {};
        v8f Dv = __builtin_amdgcn_wmma_f32_16x16x4_f32(
            /*neg_a=*/false, A, /*neg_b=*/false, Bm,
            /*c_mod=*/(short)0, C, /*reuse_a=*/false, /*reuse_b=*/false);
#pragma unroll
        for (int r = 0; r < 8; ++r)          // D layout: VGPR r -> row M = r + 8*half
            p[(size_t)r * SEQ + jt * 16] = Dv[r];
    }
}

extern "C" void kernel_launch(void* const* d_in, const int* in_sizes, int n_in,
                              void* d_out, int out_size, void* d_ws, size_t ws_size,
                              hipStream_t stream) {
    const float* head   = (const float*)d_in[0];   // 16*1024*768
    const float* dep    = (const float*)d_in[1];   // 16*1024*768
    const float* edge_W = (const float*)d_in[2];   // 1536
    const float* edge_b = (const float*)d_in[3];   // 1
    float* out = (float*)d_out;                    // 16*1024*1024

    float* sH = (float*)d_ws;                      // 16*1024 floats
    float* sD = sH + 16 * SEQ;                     // 16*1024 floats (bias folded in)

    // 2 * 16 * 1024 rows, one wave each, 8 waves per block -> 4096 blocks
    AffineEdge_rowdot<<<4096, 256, 0, stream>>>(head, dep, edge_W, edge_b, sH, sD);
    // 4096 wave-tasks, 8 waves per block -> 512 blocks
    AffineEdge_tile<<<512, 256, 0, stream>>>(sH, sD, out);
}